// GraphLoss_50508815401147
// MI455X (gfx1250) — compile-verified
//
#include <hip/hip_runtime.h>
#include <math.h>

// Problem shape (fixed by the reference's setup_inputs).
constexpr int B = 16;
constexpr int T = 2048;
constexpr int V = 512;
constexpr int L = 256;
constexpr int S = 2 * L + 1;          // 513 CTC states
#define NEGINF (-1000000000.0f)

#if defined(__has_builtin)
#  if __has_builtin(__builtin_amdgcn_global_load_async_to_lds_b32)
#    define HAVE_ASYNC_LDS 1
#  endif
#endif
#ifndef HAVE_ASYNC_LDS
#  define HAVE_ASYNC_LDS 0
#endif

// logaddexp matching jnp.logaddexp semantics for our value range.
__device__ __forceinline__ float lae(float a, float b) {
    float mx = fmaxf(a, b);
    float mn = fminf(a, b);
    return mx + log1pf(expf(mn - mx));
}

// ---------------------------------------------------------------------------
// CTC forward (numerator). One 512-thread block (16 wave32) per b.
// Thread tid owns state s = tid; thread 0 also owns state 512.
// alpha double-buffered in LDS (state s at idx s+2, 2-entry NEGINF pad),
// own-state alpha carried in a register. Emission rows staged into LDS:
//  - async path: GLOBAL_LOAD_ASYNC_TO_LDS_B32 into a 4-buffer ring, issued
//    2 steps ahead; s_wait_asynccnt 0x2 => row t landed while t+1,t+2 fly.
//    Prologue issues rows 0..2 FIRST so the DMA overlaps topology setup.
//  - fallback:   depth-1 register staging (global->VGPR->ds_store).
// Exactly 2 barriers per step.
// ---------------------------------------------------------------------------
__global__ void __launch_bounds__(512) num_kernel(const float* __restrict__ lp,
                                                  const int* __restrict__ lens,
                                                  const int* __restrict__ targets,
                                                  const int* __restrict__ tlens,
                                                  float* __restrict__ numw) {
#if HAVE_ASYNC_LDS
    __shared__ float rowbuf[4][V];      // 8 KB ring, depth-2 async pipeline
#else
    __shared__ float rowbuf[1][V];
#endif
    __shared__ float A[2][S + 7];       // state s at index s+2; idx 0,1 = NEGINF pad

    const int b   = blockIdx.x;
    const int tid = threadIdx.x;
    const float* lpb = lp + (size_t)b * T * V;

#if HAVE_ASYNC_LDS
    typedef __attribute__((address_space(1))) int GInt;    // global (prints __device__)
    typedef __attribute__((address_space(3))) int LInt;    // LDS
    #define ROW_ASYNC(t, buf)                                                   \
        __builtin_amdgcn_global_load_async_to_lds_b32(                          \
            (GInt*)(lpb + (size_t)(t) * V + tid),                               \
            (LInt*)&rowbuf[(buf)][tid], 0, 0)

    // ---- issue rows 0..2 immediately: DMA overlaps the setup below ----
    ROW_ASYNC(0, 0);
    ROW_ASYNC(1, 1);
    ROW_ASYNC(2, 2);
#endif

    const int*   tg  = targets + b * L;
    const int    len = lens[b];

    // Per-thread static CTC topology.
    const int  s    = tid;
    const int  tgt  = (s & 1) ? tg[s >> 1] : 0;            // ext[s]
    const bool skip = (s & 1) && (s >= 3) && (tgt != tg[(s >> 1) - 1]);

    float a_self = NEGINF;              // alpha for own state
    float a_top  = NEGINF;              // state 512 (thread 0 only)
    int   cur    = 0;

#if HAVE_ASYNC_LDS
    if (tid == 0) { A[0][0] = A[0][1] = A[1][0] = A[1][1] = NEGINF; }
    asm volatile("s_wait_asynccnt 0x2" ::: "memory");      // row 0 landed
    __syncthreads();
    a_self = (s < 2) ? rowbuf[0][tgt] : NEGINF;             // alpha0 from row 0
    A[cur][s + 2] = a_self;
    if (tid == 0) A[cur][S + 1] = a_top;                    // idx 514 = state 512

    for (int t = 1; t < len; ++t) {
        if (t + 2 < len) {
            ROW_ASYNC(t + 2, (t + 2) & 3);                  // reuses buffer read at t-2
            asm volatile("s_wait_asynccnt 0x2" ::: "memory");
        } else if (t + 1 < len) {
            asm volatile("s_wait_asynccnt 0x1" ::: "memory");
        } else {
            asm volatile("s_wait_asynccnt 0x0" ::: "memory");
        }
        __syncthreads();                  // row t + A[cur] visible to all waves

        const float* rb = rowbuf[t & 3];
        const int nxt = cur ^ 1;
        float e  = rb[tgt];                                  // emit gather (LDS)
        float a1 = A[cur][s + 1];
        float a2 = skip ? A[cur][s] : NEGINF;
        float na = e + lae(lae(a_self, a1), a2);
        A[nxt][s + 2] = na;
        a_self = na;

        if (tid == 0) {                                      // extra state s = 512
            float e2 = rb[0];                                // blank
            float b1 = A[cur][S];                            // state 511
            float nt = e2 + lae(lae(a_top, b1), NEGINF);
            A[nxt][S + 1] = nt;
            a_top = nt;
        }
        __syncthreads();                  // all reads done before next issue/writes
        cur = nxt;
    }
    #undef ROW_ASYNC
#else
    // ---- fallback: depth-1 register staging ----
    rowbuf[0][tid] = lpb[tid];                              // row 0, coalesced
    if (tid == 0) { A[0][0] = A[0][1] = A[1][0] = A[1][1] = NEGINF; }
    __syncthreads();
    a_self = (s < 2) ? rowbuf[0][tgt] : NEGINF;             // alpha0
    A[cur][s + 2] = a_self;
    if (tid == 0) A[cur][S + 1] = a_top;

    float pre = lpb[V + tid];                               // stage row 1

    for (int t = 1; t < len; ++t) {
        __syncthreads();                  // prior rowbuf/alpha reads complete
        rowbuf[0][tid] = pre;
        __syncthreads();                  // rowbuf(t) ready; A[cur] visible

        if (t + 1 < len) pre = lpb[(size_t)(t + 1) * V + tid];
        if (t + 2 < T)   __builtin_prefetch(lpb + (size_t)(t + 2) * V + tid, 0, 3);

        const int nxt = cur ^ 1;
        float e  = rowbuf[0][tgt];
        float a1 = A[cur][s + 1];
        float a2 = skip ? A[cur][s] : NEGINF;
        float na = e + lae(lae(a_self, a1), a2);
        A[nxt][s + 2] = na;
        a_self = na;

        if (tid == 0) {
            float e2 = rowbuf[0][0];
            float b1 = A[cur][S];
            float nt = e2 + lae(lae(a_top, b1), NEGINF);
            A[nxt][S + 1] = nt;
            a_top = nt;
        }
        cur = nxt;
    }
    __syncthreads();
#endif

    if (tid == 0) {
        const int tl = tlens[b];
        numw[b] = lae(A[cur][2 * tl + 2], A[cur][2 * tl + 1]);
    }
}

// ---------------------------------------------------------------------------
// Denominator frames. One wave32 per (b,t) row of V=512 floats.
// 4x global_load_b128 per lane, fixed-order xor-shuffle reductions.
// Bandwidth-bound; also warms L2 (192 MB holds all 67 MB) for num_kernel.
// ---------------------------------------------------------------------------
__global__ void __launch_bounds__(256) den_kernel(const float* __restrict__ lp,
                                                  float* __restrict__ frame) {
    const int row  = (blockIdx.x * 256 + threadIdx.x) >> 5;   // (b*T + t)
    const int lane = threadIdx.x & 31;
    const float4* p4 = (const float4*)(lp + (size_t)row * V);

    float4 x0 = p4[lane];
    float4 x1 = p4[lane + 32];
    float4 x2 = p4[lane + 64];
    float4 x3 = p4[lane + 96];

    float m = fmaxf(fmaxf(fmaxf(x0.x, x0.y), fmaxf(x0.z, x0.w)),
                    fmaxf(fmaxf(x1.x, x1.y), fmaxf(x1.z, x1.w)));
    m = fmaxf(m, fmaxf(fmaxf(fmaxf(x2.x, x2.y), fmaxf(x2.z, x2.w)),
                       fmaxf(fmaxf(x3.x, x3.y), fmaxf(x3.z, x3.w))));
    #pragma unroll
    for (int o = 16; o > 0; o >>= 1) m = fmaxf(m, __shfl_xor(m, o, 32));

    float s = expf(x0.x - m) + expf(x0.y - m) + expf(x0.z - m) + expf(x0.w - m)
            + expf(x1.x - m) + expf(x1.y - m) + expf(x1.z - m) + expf(x1.w - m)
            + expf(x2.x - m) + expf(x2.y - m) + expf(x2.z - m) + expf(x2.w - m)
            + expf(x3.x - m) + expf(x3.y - m) + expf(x3.z - m) + expf(x3.w - m);
    #pragma unroll
    for (int o = 16; o > 0; o >>= 1) s += __shfl_xor(s, o, 32);

    if (lane == 0) frame[row] = m + logf(s);
}

// ---------------------------------------------------------------------------
// Final combine. 16 wave32 (one per b): fixed-order masked sum of
// frame[b, t<len], then loss = mean_b (den_b - num_b) / tl_b. Deterministic.
// ---------------------------------------------------------------------------
__global__ void __launch_bounds__(512) final_kernel(const float* __restrict__ frame,
                                                    const float* __restrict__ numw,
                                                    const int* __restrict__ lens,
                                                    const int* __restrict__ tlens,
                                                    float* __restrict__ out) {
    __shared__ float parts[B];
    const int tid  = threadIdx.x;
    const int w    = tid >> 5;            // batch index
    const int lane = tid & 31;
    const int len  = lens[w];
    const float* fb = frame + (size_t)w * T;

    float s = 0.0f;
    for (int t = lane; t < T; t += 32)
        if (t < len) s += fb[t];
    #pragma unroll
    for (int o = 16; o > 0; o >>= 1) s += __shfl_xor(s, o, 32);

    if (lane == 0) {
        float tl = (float)tlens[w];
        parts[w] = (s - numw[w]) / tl;    // (den - num)/tl
    }
    __syncthreads();
    if (tid == 0) {
        float acc = 0.0f;
        #pragma unroll
        for (int i = 0; i < B; ++i) acc += parts[i];
        out[0] = acc / (float)B;
    }
}

extern "C" void kernel_launch(void* const* d_in, const int* in_sizes, int n_in,
                              void* d_out, int out_size, void* d_ws, size_t ws_size,
                              hipStream_t stream) {
    const float* lp      = (const float*)d_in[0];   // (B,T,V) f32
    const int*   lens    = (const int*)  d_in[1];   // (B,)    i32
    const int*   targets = (const int*)  d_in[2];   // (B,L)   i32
    const int*   tlens   = (const int*)  d_in[3];   // (B,)    i32

    float* frame = (float*)d_ws;                    // B*T floats
    float* numw  = frame + (size_t)B * T;           // B floats

    den_kernel  <<<(B * T) / 8, 256, 0, stream>>>(lp, frame);
    num_kernel  <<<B,           512, 0, stream>>>(lp, lens, targets, tlens, numw);
    final_kernel<<<1,           512, 0, stream>>>(frame, numw, lens, tlens, (float*)d_out);
}